// PWLNormalizor_60181081752104
// MI455X (gfx1250) — compile-verified
//
#include <hip/hip_runtime.h>

// PWL monotonic interp, (N,C=16) f32, B=32 knots/channel.
// Memory-bound: 256MB traffic -> ~11us floor @ 23.3 TB/s. No matmul -> no WMMA.
// CDNA5 features used: async global->LDS table staging (ASYNCcnt),
// wave32 bitonic sort / scan for param prep, NT b128 streaming.

#define NCH 16
#define NKNOT 32
#define TS 36                        // padded channel stride (bank-conflict break, 16B-aligned)
#define TAB_FLOATS (3 * NCH * TS)    // xs | ys | slope  = 1728 floats (6912 B)

typedef float v4f __attribute__((ext_vector_type(4)));
typedef int   v4i __attribute__((vector_size(16)));   // matches builtin prototype

// One wave (32 lanes) prepares one channel:
//  - bitonic sort of the 32 raw breakpoints across lanes
//  - inclusive wave scan of |dy| -> monotone knot values ys
//  - per-segment slope (lane = left knot index)
__device__ __forceinline__ void compute_channel_tables(
    int c, int lane,
    const float* __restrict__ xp, const float* __restrict__ y0,
    const float* __restrict__ dy,
    float& xs_o, float& ys_o, float& sl_o)
{
  float v = xp[c * NKNOT + lane];
  // Bitonic sort, 32 elements across a wave32
#pragma unroll
  for (int k = 2; k <= 32; k <<= 1) {
#pragma unroll
    for (int j = k >> 1; j >= 1; j >>= 1) {
      float other = __shfl_xor(v, j, 32);
      bool up    = ((lane & k) == 0);
      bool lower = ((lane & j) == 0);
      float lo = fminf(v, other);
      float hi = fmaxf(v, other);
      v = (up == lower) ? lo : hi;
    }
  }
  // ys[l] = y0 + sum_{j<l} |dy[j]|  (inclusive scan of shifted |dy|)
  float s = (lane >= 1) ? fabsf(dy[c * (NKNOT - 1) + (lane - 1)]) : 0.0f;
#pragma unroll
  for (int off = 1; off < 32; off <<= 1) {
    float t = __shfl_up(s, off, 32);
    if (lane >= off) s += t;
  }
  float ys = y0[c] + s;
  float xn = __shfl_down(v, 1, 32);
  float yn = __shfl_down(ys, 1, 32);
  float slope = (yn - ys) / (xn - v);  // lane 31: unused (seg <= 30)
  xs_o = v;
  ys_o = ys;
  sl_o = slope;
}

__global__ __launch_bounds__(512) void pwl_prepare(
    const float* __restrict__ xp, const float* __restrict__ y0,
    const float* __restrict__ dy, float* __restrict__ tab)
{
  const int lane = threadIdx.x & 31;
  const int c    = threadIdx.x >> 5;     // 16 waves -> 16 channels
  float xs, ys, sl;
  compute_channel_tables(c, lane, xp, y0, dy, xs, ys, sl);
  tab[0 * NCH * TS + c * TS + lane] = xs;
  tab[1 * NCH * TS + c * TS + lane] = ys;
  tab[2 * NCH * TS + c * TS + lane] = sl;
}

// Streaming eval: 1 float4 per iter, channels c0..c0+3 (c0 = 4i & 15, no wrap).
// searchsorted(left) == count of xs < xq, via 5-step branchless LDS binary search.
__device__ __forceinline__ void pwl_apply_loop(
    const float* __restrict__ x, float* __restrict__ out, int n4,
    const float* stab)
{
  const v4f* __restrict__ x4 = (const v4f*)x;
  v4f* __restrict__ o4 = (v4f*)out;
  int i = blockIdx.x * blockDim.x + threadIdx.x;
  const int stride = gridDim.x * blockDim.x;
  for (; i < n4; i += stride) {
    v4f q = __builtin_nontemporal_load(&x4[i]);
    const int c0 = (i * 4) & (NCH - 1);
    v4f r;
#pragma unroll
    for (int e = 0; e < 4; ++e) {
      const int c = c0 + e;
      const float* xsrow = stab + c * TS;
      const float xq = q[e];
      int idx = 0;
#pragma unroll
      for (int stp = 16; stp >= 1; stp >>= 1)
        idx += (xsrow[idx + stp - 1] < xq) ? stp : 0;
      int seg = idx - 1;                              // clip(idx,1,31)-1
      seg = seg < 0 ? 0 : (seg > NKNOT - 2 ? NKNOT - 2 : seg);
      const float x0 = xsrow[seg];
      const float yv = stab[1 * NCH * TS + c * TS + seg];
      const float sl = stab[2 * NCH * TS + c * TS + seg];
      r[e] = fmaf(sl, xq - x0, yv);                   // y0 + slope*(xq - x0)
    }
    __builtin_nontemporal_store(r, &o4[i]);
  }
}

__global__ __launch_bounds__(256) void pwl_forward(
    const float* __restrict__ x, const float* __restrict__ tab,
    float* __restrict__ out, int n4)
{
  __shared__ float stab[TAB_FLOATS];
#if __has_builtin(__builtin_amdgcn_global_load_async_to_lds_b128)
  // CDNA5 async copy: 432 x 16B table chunks, tracked by ASYNCcnt.
  for (int i = threadIdx.x; i < TAB_FLOATS / 4; i += blockDim.x) {
    __builtin_amdgcn_global_load_async_to_lds_b128(
        (__attribute__((address_space(1))) v4i*)(tab + 4 * i),
        (__attribute__((address_space(3))) v4i*)(stab + 4 * i),
        0, 0);
  }
#if __has_builtin(__builtin_amdgcn_s_wait_asynccnt)
  __builtin_amdgcn_s_wait_asynccnt(0);
#else
  asm volatile("s_wait_asynccnt 0" ::: "memory");
#endif
#else
  for (int i = threadIdx.x; i < TAB_FLOATS; i += blockDim.x)
    stab[i] = tab[i];
#endif
  __syncthreads();
  pwl_apply_loop(x, out, n4, stab);
}

// Fallback if d_ws is too small: each block builds its own tables (cheap: ~50 VALU
// ops per thread amortized over thousands of elements).
__global__ __launch_bounds__(256) void pwl_fused(
    const float* __restrict__ x, const float* __restrict__ xp,
    const float* __restrict__ y0, const float* __restrict__ dy,
    float* __restrict__ out, int n4)
{
  __shared__ float stab[TAB_FLOATS];
  const int lane = threadIdx.x & 31;
  const int wave = threadIdx.x >> 5;
  const int nwaves = blockDim.x >> 5;
  for (int c = wave; c < NCH; c += nwaves) {
    float xs, ys, sl;
    compute_channel_tables(c, lane, xp, y0, dy, xs, ys, sl);
    stab[0 * NCH * TS + c * TS + lane] = xs;
    stab[1 * NCH * TS + c * TS + lane] = ys;
    stab[2 * NCH * TS + c * TS + lane] = sl;
  }
  __syncthreads();
  pwl_apply_loop(x, out, n4, stab);
}

extern "C" void kernel_launch(void* const* d_in, const int* in_sizes, int n_in,
                              void* d_out, int out_size, void* d_ws, size_t ws_size,
                              hipStream_t stream) {
  const float* x  = (const float*)d_in[0];   // (N, C)
  const float* xp = (const float*)d_in[1];   // (C, B)
  const float* y0 = (const float*)d_in[2];   // (C,)
  const float* dy = (const float*)d_in[3];   // (C, B-1)
  float* out = (float*)d_out;

  const int n  = in_sizes[0];                // N*C (divisible by 4: C=16)
  const int n4 = n >> 2;

  const int threads = 256;
  int blocks = (n4 + threads - 1) / threads;
  if (blocks > 8192) blocks = 8192;          // grid-stride; ~4 float4/thread

  if (ws_size >= (size_t)(TAB_FLOATS * sizeof(float))) {
    float* tab = (float*)d_ws;
    pwl_prepare<<<1, 512, 0, stream>>>(xp, y0, dy, tab);
    pwl_forward<<<blocks, threads, 0, stream>>>(x, tab, out, n4);
  } else {
    pwl_fused<<<blocks, threads, 0, stream>>>(x, xp, y0, dy, out, n4);
  }
}